// InterpolationPredictor_14851996909739
// MI455X (gfx1250) — compile-verified
//
#include <hip/hip_runtime.h>

// Native clang vector so __builtin_nontemporal_load accepts it (HIP's float2
// is a class type and is rejected by the builtin).
typedef float v2f __attribute__((ext_vector_type(2)));

// out[r] = 2 * x[r, N-1] - x[r, N-2]
// Pure strided gather: one aligned 8-byte non-temporal load per row,
// one coalesced 4-byte store. Memory-latency bound; WMMA not applicable.
__global__ __launch_bounds__(256) void extrap_last_two_kernel(
    const float* __restrict__ x,
    float* __restrict__ out,
    int rows, int cols)
{
    int row = blockIdx.x * blockDim.x + threadIdx.x;
    if (row >= rows) return;

    // Last two elements of the row are contiguous; byte offset
    // row*cols*4 + (cols-2)*4 is 8-byte aligned for even cols (cols == 2048),
    // so this lowers to a single global_load_b64 with TH=NT.
    const v2f* p = reinterpret_cast<const v2f*>(x + (size_t)row * (size_t)cols + (size_t)(cols - 2));
    v2f v = __builtin_nontemporal_load(p);   // one-shot data: don't pollute WGP$/L2

    out[row] = 2.0f * v.y - v.x;             // single v_fma candidate

}

extern "C" void kernel_launch(void* const* d_in, const int* in_sizes, int n_in,
                              void* d_out, int out_size, void* d_ws, size_t ws_size,
                              hipStream_t stream)
{
    const float* x = (const float*)d_in[0];
    float* out = (float*)d_out;

    const int rows = out_size;                 // 131072
    const int cols = in_sizes[0] / out_size;   // 2048

    const int block = 256;                     // 8 wave32 waves per block
    const int grid  = (rows + block - 1) / block;

    extrap_last_two_kernel<<<grid, block, 0, stream>>>(x, out, rows, cols);
}